// ScoreMatching_62431644615189
// MI455X (gfx1250) — compile-verified
//
#include <hip/hip_runtime.h>

typedef __attribute__((ext_vector_type(2))) float v2f;
typedef __attribute__((ext_vector_type(8))) float v8f;

#define HH 512
#define DD 64
#define RS 516   // padded LDS row stride for R [64][RS]

// One reverse-Jacobian sweep R <- (R @ Wm) * diag(mk), fully expanded so the
// compiler keeps Wm in the global address space (global_load, LOADcnt only)
// and mk in LDS (ds_load). M=64 (4 tiles), N=512 (2 tiles/wave), K=512 step 4.
#define JAC_SWEEP(Wm, mk)                                                      \
    do {                                                                       \
        v8f acc[4][2];                                                         \
        _Pragma("unroll")                                                      \
        for (int mt = 0; mt < 4; ++mt) {                                       \
            acc[mt][0] = (v8f){0.f,0.f,0.f,0.f,0.f,0.f,0.f,0.f};               \
            acc[mt][1] = (v8f){0.f,0.f,0.f,0.f,0.f,0.f,0.f,0.f};               \
        }                                                                      \
        _Pragma("unroll 2")                                                    \
        for (int k0 = 0; k0 < HH; k0 += 4) {                                   \
            v2f afr[4];                                                        \
            _Pragma("unroll")                                                  \
            for (int mt = 0; mt < 4; ++mt) {                                   \
                const int row = mt * 16 + jn;                                  \
                afr[mt] = *(const v2f*)&Rm[row * RS + k0 + hk];                \
            }                                                                  \
            const int kb = (k0 + hk) * HH;  /* 32-bit voffset from scalar base */ \
            v2f bfr0, bfr1;                                                    \
            bfr0.x = (Wm)[kb + j0];      bfr0.y = (Wm)[kb + HH + j0];          \
            bfr1.x = (Wm)[kb + j1];      bfr1.y = (Wm)[kb + HH + j1];          \
            _Pragma("unroll")                                                  \
            for (int mt = 0; mt < 4; ++mt) {                                   \
                acc[mt][0] = __builtin_amdgcn_wmma_f32_16x16x4_f32(            \
                    false, afr[mt], false, bfr0, (short)0, acc[mt][0], false, false); \
                acc[mt][1] = __builtin_amdgcn_wmma_f32_16x16x4_f32(            \
                    false, afr[mt], false, bfr1, (short)0, acc[mt][1], false, false); \
            }                                                                  \
        }                                                                      \
        __syncthreads();  /* all reads of R complete before overwrite */       \
        const float mv0 = (mk)[j0];                                            \
        const float mv1 = (mk)[j1];                                            \
        const int rbase = (lane >> 4) << 3;                                    \
        _Pragma("unroll")                                                      \
        for (int mt = 0; mt < 4; ++mt) {                                       \
            _Pragma("unroll")                                                  \
            for (int r = 0; r < 8; ++r) {                                      \
                const int drow = mt * 16 + rbase + r;                          \
                Rm[drow * RS + j0] = acc[mt][0][r] * mv0;                      \
                Rm[drow * RS + j1] = acc[mt][1][r] * mv1;                      \
            }                                                                  \
        }                                                                      \
        __syncthreads();                                                       \
    } while (0)

// One workgroup per sample. 512 threads = 16 waves (wave32).
__global__ __launch_bounds__(512, 1)
void score_fused(const float* __restrict__ x,
                 const float* __restrict__ W1, const float* __restrict__ b1,
                 const float* __restrict__ W2, const float* __restrict__ b2,
                 const float* __restrict__ W3, const float* __restrict__ b3,
                 const float* __restrict__ W4, const float* __restrict__ b4,
                 float* __restrict__ out)
{
    extern __shared__ float lds[];
    float* Rm  = lds;               // [64][RS] tangent matrix
    float* xs  = Rm  + 64 * RS;     // [64]
    float* hs  = xs  + 64;          // [512] activations (h1 -> h2 -> h3)
    float* m1s = hs  + HH;          // [512]
    float* m2s = m1s + HH;          // [512]
    float* m3s = m2s + HH;          // [512]
    float* red = m3s + HH;          // [2] : {sum s^2, div}

    const int t    = threadIdx.x;
    const int b    = blockIdx.x;
    const int lane = t & 31;
    const int w    = t >> 5;        // wave id 0..15
    const int jn   = lane & 15;     // N / M position within a 16-wide tile
    const int hk   = (lane >> 4) << 1;  // K sub-offset: 0 or 2
    const int j0   = (2 * w) * 16 + jn; // first owned output column
    const int j1   = j0 + 16;           // second owned output column

    if (t < DD) xs[t] = x[b * DD + t];
    if (t == 0) { red[0] = 0.0f; red[1] = 0.0f; }
    __syncthreads();

    // ---------------- forward pass (VALU GEMV, one neuron per thread) -------
    // layer 1: a1 = W1 @ x + b1
    {
        float acc = b1[t];
        const float* wr = W1 + t * DD;
        #pragma unroll 8
        for (int d = 0; d < DD; ++d) acc = fmaf(wr[d], xs[d], acc);
        m1s[t] = acc > 0.0f ? 1.0f : 0.0f;
        hs[t]  = acc > 0.0f ? acc : 0.0f;
    }
    __syncthreads();
    // layer 2
    {
        float acc = b2[t];
        const float* wr = W2 + t * HH;
        #pragma unroll 4
        for (int j = 0; j < HH; ++j) acc = fmaf(wr[j], hs[j], acc);
        __syncthreads();
        m2s[t] = acc > 0.0f ? 1.0f : 0.0f;
        hs[t]  = acc > 0.0f ? acc : 0.0f;
    }
    __syncthreads();
    // layer 3
    {
        float acc = b3[t];
        const float* wr = W3 + t * HH;
        #pragma unroll 4
        for (int j = 0; j < HH; ++j) acc = fmaf(wr[j], hs[j], acc);
        __syncthreads();
        m3s[t] = acc > 0.0f ? 1.0f : 0.0f;
        hs[t]  = acc > 0.0f ? acc : 0.0f;   // hs now holds h3 (kept)
    }
    __syncthreads();

    // s = W4 @ h3 + b4 ; accumulate sum s^2 (0.5 factor applied at the end)
    if (t < DD) {
        float acc = b4[t];
        const float* wr = W4 + t * HH;
        #pragma unroll 4
        for (int j = 0; j < HH; ++j) acc = fmaf(wr[j], hs[j], acc);
        atomicAdd(&red[0], acc * acc);
    }

    // ---------------- R0 = W4 * diag(m3)   [64 x 512] -----------------------
    for (int idx = t; idx < DD * HH; idx += 512) {
        const int d = idx >> 9;
        const int i = idx & (HH - 1);
        Rm[d * RS + i] = W4[idx] * m3s[i];
    }
    __syncthreads();

    // ---------------- two reverse GEMM sweeps on the WMMA fp32 pipe ---------
    JAC_SWEEP(W3, m2s);   // R1 = (R0 @ W3) * diag(m2)
    JAC_SWEEP(W2, m1s);   // R2 = (R1 @ W2) * diag(m1)

    // ---------------- div = sum_{d,k} R2[d,k] * W1[k,d] ---------------------
    {
        const float* wr = W1 + t * DD;   // W1 row k = t, contiguous over d
        float loc = 0.0f;
        #pragma unroll 8
        for (int d = 0; d < DD; ++d) loc = fmaf(Rm[d * RS + t], wr[d], loc);
        atomicAdd(&red[1], loc);
    }
    __syncthreads();

    if (t == 0) out[b] = 0.5f * red[0] + red[1];
}

extern "C" void kernel_launch(void* const* d_in, const int* in_sizes, int n_in,
                              void* d_out, int out_size, void* d_ws, size_t ws_size,
                              hipStream_t stream) {
    (void)in_sizes; (void)n_in; (void)d_ws; (void)ws_size; (void)out_size;
    const float* x  = (const float*)d_in[0];
    const float* W1 = (const float*)d_in[1];
    const float* b1 = (const float*)d_in[2];
    const float* W2 = (const float*)d_in[3];
    const float* b2 = (const float*)d_in[4];
    const float* W3 = (const float*)d_in[5];
    const float* b3 = (const float*)d_in[6];
    const float* W4 = (const float*)d_in[7];
    const float* b4 = (const float*)d_in[8];
    float* out = (float*)d_out;

    const size_t shbytes = (size_t)(64 * RS + 64 + 4 * HH + 16) * sizeof(float);
    score_fused<<<dim3(2048), dim3(512), shbytes, stream>>>(
        x, W1, b1, W2, b2, W3, b3, W4, b4, out);
}